// GraphConvHeteroBlock_72095321030697
// MI455X (gfx1250) — compile-verified
//
#include <hip/hip_runtime.h>
#include <math.h>

#define Nn 8192
#define Dd 128
#define Hh 2
#define Rr 15
#define Ee 16384
#define Ll 2

__constant__ int c_SRC[Rr] = {0,2,1,0,2,1,0,2,1,0,0,2,2,1,1};
__constant__ int c_DST[Rr] = {0,2,1,0,2,1,0,2,1,2,1,0,1,0,2};

typedef float v2f __attribute__((ext_vector_type(2)));
typedef float v8f __attribute__((ext_vector_type(8)));

__device__ __forceinline__ v8f wmma_f32(v2f a, v2f b, v8f c) {
  // D = A(16x4,f32) * B(4x16,f32) + C(16x16,f32)
  return __builtin_amdgcn_wmma_f32_16x16x4_f32(false, a, false, b, (short)0, c, false, false);
}

__device__ __forceinline__ void atomicMaxFloat(float* addr, float val) {
  if (val >= 0.0f) atomicMax((int*)addr, __float_as_int(val));
  else             atomicMin((unsigned int*)addr, __float_as_uint(val));
}

// ---------------- utility ----------------
__global__ void fill_kernel(float* __restrict__ p, float v, int n) {
  int i = blockIdx.x * blockDim.x + threadIdx.x;
  if (i < n) p[i] = v;
}

__global__ void relu_scale_kernel(const float* __restrict__ xn, float* __restrict__ x, int n) {
  int i = blockIdx.x * blockDim.x + threadIdx.x;
  if (i < n) x[i] = fmaxf(xn[i] * 0.2f, 0.0f);
}

// ---------------- RGCN message scatter (per relation) ----------------
__global__ void scatter_msg_kernel(const float* __restrict__ x, const int* __restrict__ edges,
                                   float* __restrict__ msg, int r, int srcType) {
  int eg = blockIdx.x * (blockDim.x >> 5) + (threadIdx.x >> 5);
  if (eg >= Ee) return;
  int lane = threadIdx.x & 31;
  const int* ep = edges + (size_t)r * 2 * Ee;
  int s = ep[eg];
  int d = ep[Ee + eg];
  const float4* xs = (const float4*)(x + ((size_t)srcType * Nn + s) * Dd);
  float4 v = xs[lane];
  float* md = msg + (size_t)d * Dd + lane * 4;
  atomicAdd(md + 0, v.x); atomicAdd(md + 1, v.y);
  atomicAdd(md + 2, v.z); atomicAdd(md + 3, v.w);
}

// ---------------- dual GEMM: C += msg@Wrel + xdst@Wroot + brel ----------------
// A1,A2: Nn x 128 row-major.  B1,B2: 128 x 128 row-major.  C: Nn x 128, += (stream-serialized).
__global__ __launch_bounds__(128) void rgcn_gemm_kernel(
    const float* __restrict__ A1, const float* __restrict__ A2,
    const float* __restrict__ B1, const float* __restrict__ B2,
    const float* __restrict__ bias, float* __restrict__ C) {
  const int lane = threadIdx.x & 31;
  const int wave = threadIdx.x >> 5;
  const int n0 = blockIdx.y * 16;
  const int m_base = blockIdx.x * 256 + wave * 64;
  const int nn = lane & 15;
  const int h  = lane >> 4;
  // Preload full-K B strips in registers (B layout: vgpr j, lane-half h -> row 4k+2h+j, col nn)
  v2f b1s[32], b2s[32];
#pragma unroll
  for (int k = 0; k < 32; ++k) {
    int kr = 4 * k + 2 * h;
    b1s[k][0] = B1[(size_t)(kr + 0) * Dd + n0 + nn];
    b1s[k][1] = B1[(size_t)(kr + 1) * Dd + n0 + nn];
    b2s[k][0] = B2[(size_t)(kr + 0) * Dd + n0 + nn];
    b2s[k][1] = B2[(size_t)(kr + 1) * Dd + n0 + nn];
  }
  const float bv = bias[n0 + nn];
  for (int mt = 0; mt < 4; ++mt) {
    const int m0 = m_base + mt * 16;
    const float* a1p = A1 + (size_t)(m0 + nn) * Dd + 2 * h; // A layout: row nn, K = 2h+j
    const float* a2p = A2 + (size_t)(m0 + nn) * Dd + 2 * h;
    v8f acc;
#pragma unroll
    for (int j = 0; j < 8; ++j) acc[j] = bv;
#pragma unroll
    for (int k = 0; k < 32; ++k) {
      v2f a1 = *(const v2f*)(a1p + 4 * k);
      acc = wmma_f32(a1, b1s[k], acc);
      v2f a2 = *(const v2f*)(a2p + 4 * k);
      acc = wmma_f32(a2, b2s[k], acc);
    }
    float* cp = C + (size_t)m0 * Dd + n0 + nn;
#pragma unroll
    for (int j = 0; j < 8; ++j) {
      int m = j + 8 * h; // C layout: vgpr j, lane-half h -> row j+8h, col nn
      cp[(size_t)m * Dd] += acc[j];
    }
  }
}

// ---------------- linear: C = A(Mx128) @ B(128xNc) + bias (overwrite) ----------------
__global__ __launch_bounds__(128) void linear_kernel(
    const float* __restrict__ A, const float* __restrict__ B,
    const float* __restrict__ bias, float* __restrict__ C, int ld) {
  const int lane = threadIdx.x & 31;
  const int wave = threadIdx.x >> 5;
  const int n0 = blockIdx.y * 16;
  const int m_base = blockIdx.x * 256 + wave * 64;
  const int nn = lane & 15;
  const int h  = lane >> 4;
  v2f bs[32];
#pragma unroll
  for (int k = 0; k < 32; ++k) {
    int kr = 4 * k + 2 * h;
    bs[k][0] = B[(size_t)(kr + 0) * ld + n0 + nn];
    bs[k][1] = B[(size_t)(kr + 1) * ld + n0 + nn];
  }
  const float bv = bias[n0 + nn];
  for (int mt = 0; mt < 4; ++mt) {
    const int m0 = m_base + mt * 16;
    const float* ap = A + (size_t)(m0 + nn) * Dd + 2 * h;
    v8f acc;
#pragma unroll
    for (int j = 0; j < 8; ++j) acc[j] = bv;
#pragma unroll
    for (int k = 0; k < 32; ++k) {
      v2f a = *(const v2f*)(ap + 4 * k);
      acc = wmma_f32(a, bs[k], acc);
    }
    float* cp = C + (size_t)m0 * ld + n0 + nn;
#pragma unroll
    for (int j = 0; j < 8; ++j) {
      int m = j + 8 * h;
      cp[(size_t)m * ld] = acc[j];
    }
  }
}

// ---------------- attention: logits + segment max ----------------
__global__ void logits_kernel(const float* __restrict__ Q, const float* __restrict__ K,
                              const int* __restrict__ edges, float* __restrict__ P,
                              float* __restrict__ M) {
  int eg = blockIdx.x * (blockDim.x >> 5) + (threadIdx.x >> 5);
  if (eg >= Rr * Ee) return;
  int lane = threadIdx.x & 31;
  int r = eg / Ee, e = eg - r * Ee;
  const int* ep = edges + (size_t)r * 2 * Ee;
  int sh = c_SRC[r] * Nn + ep[e];
  int dh = c_DST[r] * Nn + ep[Ee + e];
  const float4* qp = (const float4*)(Q + (size_t)dh * (Hh * Dd));
  const float4* kp = (const float4*)(K + (size_t)sh * (Hh * Dd));
  float4 q0 = qp[lane],      k0 = kp[lane];
  float4 q1 = qp[lane + 32], k1 = kp[lane + 32];
  float p0 = q0.x * k0.x + q0.y * k0.y + q0.z * k0.z + q0.w * k0.w;
  float p1 = q1.x * k1.x + q1.y * k1.y + q1.z * k1.z + q1.w * k1.w;
#pragma unroll
  for (int off = 16; off > 0; off >>= 1) {
    p0 += __shfl_xor(p0, off, 32);
    p1 += __shfl_xor(p1, off, 32);
  }
  if (lane == 0) {
    const float scale = 0.08838834764831845f; // 1/sqrt(128)
    float l0 = p0 * scale, l1 = p1 * scale;
    P[(size_t)eg * 2 + 0] = l0;
    P[(size_t)eg * 2 + 1] = l1;
    atomicMaxFloat(M + (size_t)dh * 2 + 0, l0);
    atomicMaxFloat(M + (size_t)dh * 2 + 1, l1);
  }
}

// ---------------- attention: p = exp(l - m), denom ----------------
__global__ void pexp_kernel(float* __restrict__ P, const float* __restrict__ M,
                            float* __restrict__ DEN, const int* __restrict__ edges) {
  int eg = blockIdx.x * blockDim.x + threadIdx.x;
  if (eg >= Rr * Ee) return;
  int r = eg / Ee, e = eg - r * Ee;
  int dh = c_DST[r] * Nn + edges[(size_t)r * 2 * Ee + Ee + e];
  float p0 = expf(P[(size_t)eg * 2 + 0] - M[(size_t)dh * 2 + 0]);
  float p1 = expf(P[(size_t)eg * 2 + 1] - M[(size_t)dh * 2 + 1]);
  P[(size_t)eg * 2 + 0] = p0;
  P[(size_t)eg * 2 + 1] = p1;
  atomicAdd(DEN + (size_t)dh * 2 + 0, p0);
  atomicAdd(DEN + (size_t)dh * 2 + 1, p1);
}

// ---------------- attention: aggregate alpha*v (mean over heads) ----------------
__global__ void agg_kernel(const float* __restrict__ P, const float* __restrict__ DEN,
                           const float* __restrict__ V, const int* __restrict__ edges,
                           float* __restrict__ OUT) {
  int eg = blockIdx.x * (blockDim.x >> 5) + (threadIdx.x >> 5);
  if (eg >= Rr * Ee) return;
  int lane = threadIdx.x & 31;
  int r = eg / Ee, e = eg - r * Ee;
  const int* ep = edges + (size_t)r * 2 * Ee;
  int sh = c_SRC[r] * Nn + ep[e];
  int dh = c_DST[r] * Nn + ep[Ee + e];
  float a0 = 0.5f * P[(size_t)eg * 2 + 0] / DEN[(size_t)dh * 2 + 0];
  float a1 = 0.5f * P[(size_t)eg * 2 + 1] / DEN[(size_t)dh * 2 + 1];
  const float4* vp = (const float4*)(V + (size_t)sh * (Hh * Dd));
  float4 v0 = vp[lane], v1 = vp[lane + 32];
  float* op = OUT + (size_t)dh * Dd + lane * 4;
  atomicAdd(op + 0, a0 * v0.x + a1 * v1.x);
  atomicAdd(op + 1, a0 * v0.y + a1 * v1.y);
  atomicAdd(op + 2, a0 * v0.z + a1 * v1.z);
  atomicAdd(op + 3, a0 * v0.w + a1 * v1.w);
}

// ---------------- column stats (coalesced: thread = column) ----------------
__global__ void colsum_kernel(const float* __restrict__ OUT, float* __restrict__ cst) {
  int d = threadIdx.x; // 128 threads
  int row0 = blockIdx.x * 256;
  float s = 0.0f, sq = 0.0f;
  for (int i = 0; i < 256; ++i) {
    float v = OUT[(size_t)(row0 + i) * Dd + d];
    s += v; sq += v * v;
  }
  atomicAdd(cst + d, s);
  atomicAdd(cst + Dd + d, sq);
}

__global__ void finalize_stats_kernel(float* __restrict__ cst) {
  int d = threadIdx.x;
  const float inv = 1.0f / (3.0f * Nn);
  float mu = cst[d] * inv;
  float var = cst[Dd + d] * inv - mu * mu;
  cst[2 * Dd + d] = mu;
  cst[3 * Dd + d] = rsqrtf(var + 1e-5f);
}

__global__ void norm_kernel(const float* __restrict__ OUT, const float* __restrict__ cst,
                            const float* __restrict__ gamma, const float* __restrict__ beta,
                            float* __restrict__ y, int n) {
  int i = blockIdx.x * blockDim.x + threadIdx.x;
  if (i >= n) return;
  int d = i & (Dd - 1);
  float v = (OUT[i] - cst[2 * Dd + d]) * cst[3 * Dd + d] * gamma[d] + beta[d];
  y[i] = v >= 0.0f ? v : 0.01f * v;
}

// ---------------- host orchestration ----------------
extern "C" void kernel_launch(void* const* d_in, const int* in_sizes, int n_in,
                              void* d_out, int out_size, void* d_ws, size_t ws_size,
                              hipStream_t stream) {
  static const int hDST[Rr] = {0,2,1,0,2,1,0,2,1,2,1,0,1,0,2};
  static const int hSRC[Rr] = {0,2,1,0,2,1,0,2,1,0,0,2,2,1,1};

  const float* x_audio = (const float*)d_in[0];
  const float* x_text  = (const float*)d_in[1];
  const float* x_vis   = (const float*)d_in[2];
  const int*   edges   = (const int*)d_in[3];
  const float* Wrel    = (const float*)d_in[4];
  const float* brel    = (const float*)d_in[5];
  const float* Wroot   = (const float*)d_in[6];
  const float* Wq      = (const float*)d_in[7];
  const float* bq      = (const float*)d_in[8];
  const float* Wk      = (const float*)d_in[9];
  const float* bk      = (const float*)d_in[10];
  const float* Wv      = (const float*)d_in[11];
  const float* bv      = (const float*)d_in[12];
  const float* Wskip   = (const float*)d_in[13];
  const float* bskip   = (const float*)d_in[14];
  const float* gamma   = (const float*)d_in[15];
  const float* beta    = (const float*)d_in[16];
  float* y = (float*)d_out;

  float* ws = (float*)d_ws;
  const size_t ND3 = (size_t)3 * Nn * Dd;
  float* X   = ws;
  float* XN  = X + ND3;
  float* MSG = XN + ND3;
  float* Q   = MSG + (size_t)Nn * Dd;
  float* Kb  = Q + (size_t)3 * Nn * Hh * Dd;
  float* Vb  = Kb + (size_t)3 * Nn * Hh * Dd;
  float* P   = Vb + (size_t)3 * Nn * Hh * Dd;
  float* M   = P + (size_t)Rr * Ee * Hh;
  float* DEN = M + (size_t)3 * Nn * Hh;
  float* CST = DEN + (size_t)3 * Nn * Hh;

  const size_t NDb = (size_t)Nn * Dd * sizeof(float);
  hipMemcpyAsync(X,              x_audio, NDb, hipMemcpyDeviceToDevice, stream);
  hipMemcpyAsync(X + Nn * Dd,    x_text,  NDb, hipMemcpyDeviceToDevice, stream);
  hipMemcpyAsync(X + 2 * Nn * Dd,x_vis,   NDb, hipMemcpyDeviceToDevice, stream);

  auto cdiv = [](int a, int b) { return (a + b - 1) / b; };

  // ---- RGCN layers ----
  for (int l = 0; l < Ll; ++l) {
    fill_kernel<<<cdiv((int)ND3, 256), 256, 0, stream>>>(XN, 0.0f, (int)ND3);
    for (int r = 0; r < Rr; ++r) {
      fill_kernel<<<cdiv(Nn * Dd, 256), 256, 0, stream>>>(MSG, 0.0f, Nn * Dd);
      scatter_msg_kernel<<<Ee / 4, 128, 0, stream>>>(X, edges, MSG, r, hSRC[r]);
      size_t wo = ((size_t)l * Rr + r);
      rgcn_gemm_kernel<<<dim3(Nn / 256, Dd / 16), 128, 0, stream>>>(
          MSG, X + (size_t)hDST[r] * Nn * Dd,
          Wrel + wo * Dd * Dd, Wroot + wo * Dd * Dd, brel + wo * Dd,
          XN + (size_t)hDST[r] * Nn * Dd);
    }
    relu_scale_kernel<<<cdiv((int)ND3, 256), 256, 0, stream>>>(XN, X, (int)ND3);
  }

  // ---- projections (q, k, v, skip) ----
  const int Mrows = 3 * Nn;
  linear_kernel<<<dim3(Mrows / 256, (Hh * Dd) / 16), 128, 0, stream>>>(X, Wq, bq, Q,  Hh * Dd);
  linear_kernel<<<dim3(Mrows / 256, (Hh * Dd) / 16), 128, 0, stream>>>(X, Wk, bk, Kb, Hh * Dd);
  linear_kernel<<<dim3(Mrows / 256, (Hh * Dd) / 16), 128, 0, stream>>>(X, Wv, bv, Vb, Hh * Dd);
  linear_kernel<<<dim3(Mrows / 256, Dd / 16), 128, 0, stream>>>(X, Wskip, bskip, XN, Dd);

  // ---- segment softmax attention ----
  int segN = 3 * Nn * Hh;
  fill_kernel<<<cdiv(segN, 256), 256, 0, stream>>>(M, -INFINITY, segN);
  fill_kernel<<<cdiv(segN, 256), 256, 0, stream>>>(DEN, 0.0f, segN);
  fill_kernel<<<1, 256, 0, stream>>>(CST, 0.0f, 2 * Dd);

  int NE = Rr * Ee;
  logits_kernel<<<cdiv(NE, 4), 128, 0, stream>>>(Q, Kb, edges, P, M);
  pexp_kernel<<<cdiv(NE, 256), 256, 0, stream>>>(P, M, DEN, edges);
  agg_kernel<<<cdiv(NE, 4), 128, 0, stream>>>(P, DEN, Vb, edges, XN);

  // ---- layernorm over rows + leaky relu ----
  colsum_kernel<<<Mrows / 256, Dd, 0, stream>>>(XN, CST);
  finalize_stats_kernel<<<1, Dd, 0, stream>>>(CST);
  norm_kernel<<<cdiv((int)ND3, 256), 256, 0, stream>>>(XN, CST, gamma, beta, y, (int)ND3);
}